// GNNNode_6640019439760
// MI455X (gfx1250) — compile-verified
//
#include <hip/hip_runtime.h>
#include <hip/hip_bf16.h>

// ---------------- problem constants (from reference) ----------------
#define BATCH 8192
#define KFAN  16
#define VOCAB 50000
#define DIM   64
#define CR    8
#define NR    64
#define CC    16
#define NC    128
#define OUTD  32
#define KTOT  (CC * DIM + NC)         // 1152
#define OCOLS (CR * DIM + NR + OUTD)  // 608
#define EPS   1e-5f

typedef float v2f __attribute__((ext_vector_type(2)));
typedef float v8f __attribute__((ext_vector_type(8)));
typedef unsigned int v4u __attribute__((ext_vector_type(4)));
typedef int v8i __attribute__((ext_vector_type(8)));
typedef int v4i __attribute__((ext_vector_type(4)));

#if __has_builtin(__builtin_amdgcn_tensor_load_to_lds) && \
    __has_builtin(__builtin_amdgcn_s_wait_tensorcnt)
#define USE_TDM 1
#else
#define USE_TDM 0
#endif

// ---------------------------------------------------------------
// BatchNorm statistics over gathered rows: one block per column.
// Deterministic fixed-order tree reduction (no float atomics).
// Emits fused affine: scale = gamma*rsqrt(var+eps), shift = beta - mu*scale
// ---------------------------------------------------------------
__global__ void bn_stats_kernel(const int* __restrict__ sel,
                                const float* __restrict__ data,
                                int nrows, int ncols,
                                const float* __restrict__ gamma,
                                const float* __restrict__ beta,
                                float* __restrict__ scale_out,
                                float* __restrict__ shift_out) {
    const int c   = blockIdx.x;
    const int tid = threadIdx.x;
    float sum = 0.0f, sq = 0.0f;
    for (int r = tid; r < nrows; r += 256) {
        float v = data[(size_t)sel[r] * (size_t)ncols + (size_t)c];
        sum += v;
        sq  += v * v;
    }
    __shared__ float ssum[256];
    __shared__ float ssq[256];
    ssum[tid] = sum;
    ssq[tid]  = sq;
    __syncthreads();
    for (int s = 128; s > 0; s >>= 1) {
        if (tid < s) {
            ssum[tid] += ssum[tid + s];
            ssq[tid]  += ssq[tid + s];
        }
        __syncthreads();
    }
    if (tid == 0) {
        float inv_n = 1.0f / (float)nrows;
        float mu    = ssum[0] * inv_n;
        float var   = ssq[0] * inv_n - mu * mu;
        float sc    = gamma[c] * rsqrtf(var + EPS);
        scale_out[c] = sc;
        shift_out[c] = beta[c] - mu * sc;
    }
}

// ---------------------------------------------------------------
// Root path: one wave (32 lanes) per root row.
// ---------------------------------------------------------------
__global__ void root_kernel(const int* __restrict__ idx,
                            const int* __restrict__ root_cat,
                            const float* __restrict__ root_num,
                            const float* __restrict__ emb_root,
                            const float* __restrict__ scale_r,
                            const float* __restrict__ shift_r,
                            float* __restrict__ out) {
    const int lane = threadIdx.x & 31;
    const int wv   = threadIdx.x >> 5;
    const int r    = blockIdx.x * 8 + wv;
    const size_t row = (size_t)idx[r];
    float* __restrict__ orow = out + (size_t)r * OCOLS;

#pragma unroll
    for (int t = 0; t < CR; ++t) {
        int c = root_cat[row * CR + t];
        const float2* src = (const float2*)(emb_root + ((size_t)t * VOCAB + (size_t)c) * DIM);
        ((float2*)(orow + t * DIM))[lane] = src[lane];
    }
    const float2* nsrc = (const float2*)(root_num + row * NR);
    float2 v = nsrc[lane];
    int c0 = lane * 2;
    v.x = v.x * scale_r[c0]     + shift_r[c0];
    v.y = v.y * scale_r[c0 + 1] + shift_r[c0 + 1];
    ((float2*)(orow + CR * DIM))[lane] = v;
}

// ---------------------------------------------------------------
// Child path (gather -> BN -> Linear(WMMA f32) -> ReLU -> mean pool).
// Workgroup: 256 threads (8 waves), 64 consecutive child rows
//          = 4 complete mean-pool groups of K=16.
// K processed in 18 blocks of 64 (16 embedding tables + 2 numeric halves).
// Embedding H-tiles staged by the Tensor Data Mover in gather mode:
//   waves 0-3 each issue one TENSOR_LOAD_TO_LDS with 16 packed 16-bit
//   row indices; TDM pad fields reproduce the 66-float LDS stride.
// ---------------------------------------------------------------
#define LDSPAD 66   // padded row stride (floats) -> conflict-free lane reads

__global__ void child_kernel(const int* __restrict__ child_flat_idx,
                             const int* __restrict__ child_cat,
                             const float* __restrict__ child_num,
                             const float* __restrict__ emb_child,
                             const float* __restrict__ Wmat,
                             const float* __restrict__ bias,
                             const float* __restrict__ scale_c,
                             const float* __restrict__ shift_c,
                             float* __restrict__ out) {
    __shared__ float lds_h[64][LDSPAD];   // H tile: 64 rows x 64 K-cols (+2 pad)
    __shared__ float lds_w[OUTD][LDSPAD]; // W slice: 32 out x 64 K-cols
    __shared__ float s_scale[NC];
    __shared__ float s_shift[NC];

    const int tid  = threadIdx.x;
    const int lane = tid & 31;
    const int wv   = tid >> 5;
    const int mtile = wv >> 1;      // 0..3
    const int ntile = wv & 1;       // 0..1

    if (tid < NC) {
        s_scale[tid] = scale_c[tid];
        s_shift[tid] = shift_c[tid];
    }

    // gather assignment (manual paths): 4 threads per row, 16 floats each
    const int m = tid >> 2;          // 0..63
    const int q = tid & 3;           // 0..3
    const size_t g = (size_t)blockIdx.x * 64 + (size_t)m;   // child row id

    const int mrow  = mtile * 16 + (lane & 15);
    const int nrow  = ntile * 16 + (lane & 15);
    const int khalf = (lane >> 4) * 2;   // lanes 0-15 -> K{0,1}; 16-31 -> K{2,3}

#if USE_TDM
    // row id for this wave's 16-row TDM slice (lanes 16-31 mirror 0-15)
    const size_t grow = (size_t)blockIdx.x * 64 + (size_t)(wv & 3) * 16 + (size_t)(lane & 15);
    // static D# pieces (wave-uniform)
    const unsigned lds_dst = (unsigned)(size_t)(void*)&lds_h[(wv & 3) * 16][0]; // LDS byte offset (flat addr[31:0])
#endif

    v8f acc = {};   // f32 accumulators, 16x16 D tile

    for (int t = 0; t < 18; ++t) {
        __syncthreads();   // previous iteration's LDS reads done before overwrite

        // ---- stage H K-block into LDS ----
#if USE_TDM
        if (t < CC) {
            if (wv < 4) {
                // 16 gather indices for this wave's row slice
                int idxv = child_cat[grow * CC + (size_t)t];
                unsigned p0, p1, p2, p3, p4, p5, p6, p7;
                p0 = ((unsigned)__builtin_amdgcn_readlane(idxv, 0)  & 0xffffu) |
                     ((unsigned)__builtin_amdgcn_readlane(idxv, 1)  << 16);
                p1 = ((unsigned)__builtin_amdgcn_readlane(idxv, 2)  & 0xffffu) |
                     ((unsigned)__builtin_amdgcn_readlane(idxv, 3)  << 16);
                p2 = ((unsigned)__builtin_amdgcn_readlane(idxv, 4)  & 0xffffu) |
                     ((unsigned)__builtin_amdgcn_readlane(idxv, 5)  << 16);
                p3 = ((unsigned)__builtin_amdgcn_readlane(idxv, 6)  & 0xffffu) |
                     ((unsigned)__builtin_amdgcn_readlane(idxv, 7)  << 16);
                p4 = ((unsigned)__builtin_amdgcn_readlane(idxv, 8)  & 0xffffu) |
                     ((unsigned)__builtin_amdgcn_readlane(idxv, 9)  << 16);
                p5 = ((unsigned)__builtin_amdgcn_readlane(idxv, 10) & 0xffffu) |
                     ((unsigned)__builtin_amdgcn_readlane(idxv, 11) << 16);
                p6 = ((unsigned)__builtin_amdgcn_readlane(idxv, 12) & 0xffffu) |
                     ((unsigned)__builtin_amdgcn_readlane(idxv, 13) << 16);
                p7 = ((unsigned)__builtin_amdgcn_readlane(idxv, 14) & 0xffffu) |
                     ((unsigned)__builtin_amdgcn_readlane(idxv, 15) << 16);

                unsigned long long gaddr =
                    (unsigned long long)(size_t)(const void*)(emb_child + (size_t)t * VOCAB * DIM);

                v4u g0;
                g0.x = 0x80000001u;                 // count=1, gather_mode=1, 16-bit indices
                g0.y = lds_dst;                     // lds_addr (bytes)
                g0.z = (unsigned)(gaddr & 0xffffffffu);          // global_addr[31:0]
                g0.w = (unsigned)((gaddr >> 32) & 0x01ffffffu)   // global_addr[56:32]
                     | (2u << 30);                               // type=2 (image)

                v8i g1;
                g1[0] = (int)((2u << 16)            // data_size = 4B
                            | (1u << 20)            // pad_enable
                            | (5u << 22)            // pad_interval: 64 DWORDs
                            | (1u << 25));          // pad_amount: 2 DWORDs -> stride 66
                g1[1] = (int)(64u << 16);           // tensor_dim0 = 64 (row width)
                g1[2] = (int)((unsigned)VOCAB << 16); // tensor_dim1[15:0] = 50000
                g1[3] = (int)(64u << 16);           // tile_dim0 = 64
                g1[4] = 16;                         // tile_dim1 = 16 valid indices
                g1[5] = 64;                         // tensor_dim0_stride = 64 elements
                g1[6] = 0;
                g1[7] = 0;

                v4i g2, g3;
                g2[0] = (int)p0; g2[1] = (int)p1; g2[2] = (int)p2; g2[3] = (int)p3;
                g3[0] = (int)p4; g3[1] = (int)p5; g3[2] = (int)p6; g3[3] = (int)p7;

                v8i g4 = {};   // unused trailing SGPR group (VADDR4 slot) -> zeros

                __builtin_amdgcn_tensor_load_to_lds(g0, g1, g2, g3, g4, 0);
            }
        } else
#else
        if (t < CC) {
            // manual path: copy emb_child[t, cat, 0:64]
            int c = child_cat[g * CC + t];
            const float2* src = (const float2*)(emb_child + ((size_t)t * VOCAB + (size_t)c) * DIM) + q * 8;
            float2* dst = (float2*)&lds_h[m][q * 16];
#pragma unroll
            for (int i = 0; i < 8; ++i) dst[i] = src[i];
        } else
#endif
        {
            // numeric half (t==16 -> cols 0:64, t==17 -> cols 64:128), BN affine fused
            int off   = (t - CC) * 64;
            int cbase = off + q * 16;
            const float2* src = (const float2*)(child_num + g * NC + cbase);
            float2* dst = (float2*)&lds_h[m][q * 16];
#pragma unroll
            for (int i = 0; i < 8; ++i) {
                float2 v = src[i];
                int c0 = cbase + 2 * i;
                v.x = v.x * s_scale[c0]     + s_shift[c0];
                v.y = v.y * s_scale[c0 + 1] + s_shift[c0 + 1];
                dst[i] = v;
            }
        }

        // ---- stage W K-slice into LDS: lds_w[n][k] = W[n][t*64+k] ----
        {
            int n  = tid >> 3;          // 0..31
            int k8 = (tid & 7) * 8;     // 0..56
            const float* wsrc = Wmat + (size_t)n * KTOT + t * 64 + k8;
#pragma unroll
            for (int i = 0; i < 8; ++i) lds_w[n][k8 + i] = wsrc[i];
        }

#if USE_TDM
        if (t < CC && wv < 4) __builtin_amdgcn_s_wait_tensorcnt(0);
#endif
        __syncthreads();

        // ---- 16 WMMA K-steps (K=4 each) over this 64-wide K-block ----
#pragma unroll
        for (int ks = 0; ks < 16; ++ks) {
            int k0 = ks * 4 + khalf;
            v2f a  = *(const v2f*)&lds_h[mrow][k0];   // A[m][k0], A[m][k0+1]
            v2f bb = *(const v2f*)&lds_w[nrow][k0];   // B[k0][n], B[k0+1][n] (= W[n][k])
            acc = __builtin_amdgcn_wmma_f32_16x16x4_f32(
                false, a, false, bb, (short)0, acc, false, false);
        }
    }

    // ---- epilogue: bias + ReLU + mean over the 16-row group ----
    const int ncol = ntile * 16 + (lane & 15);
    const float bv = bias[ncol];
    float s = 0.0f;
#pragma unroll
    for (int j = 0; j < 8; ++j) {
        float v = acc[j] + bv;        // D vgpr j: row j + 8*(lane>=16)
        s += (v > 0.0f) ? v : 0.0f;
    }
    s += __shfl_xor(s, 16, 32);       // fold the two row-halves -> sum of 16 rows
    if (lane < 16) {
        int broot = blockIdx.x * 4 + mtile;
        out[(size_t)broot * OCOLS + (CR * DIM + NR) + ncol] = s * (1.0f / (float)KFAN);
    }
}

// ---------------------------------------------------------------
extern "C" void kernel_launch(void* const* d_in, const int* in_sizes, int n_in,
                              void* d_out, int out_size, void* d_ws, size_t ws_size,
                              hipStream_t stream) {
    const int*   idx            = (const int*)d_in[0];
    const int*   child_flat_idx = (const int*)d_in[1];
    const int*   root_cat       = (const int*)d_in[2];
    const int*   child_cat      = (const int*)d_in[3];
    const float* root_num       = (const float*)d_in[4];
    const float* child_num      = (const float*)d_in[5];
    const float* emb_root       = (const float*)d_in[6];
    const float* emb_child      = (const float*)d_in[7];
    const float* Wmat           = (const float*)d_in[8];
    const float* bias           = (const float*)d_in[9];
    const float* gamma_r        = (const float*)d_in[10];
    const float* beta_r         = (const float*)d_in[11];
    const float* gamma_c        = (const float*)d_in[12];
    const float* beta_c         = (const float*)d_in[13];
    float* out = (float*)d_out;

    float* ws      = (float*)d_ws;
    float* scale_r = ws;        // [64]
    float* shift_r = ws + 64;   // [64]
    float* scale_c = ws + 128;  // [128]
    float* shift_c = ws + 256;  // [128]

    bn_stats_kernel<<<NR, 256, 0, stream>>>(idx, root_num, BATCH, NR,
                                            gamma_r, beta_r, scale_r, shift_r);
    bn_stats_kernel<<<NC, 256, 0, stream>>>(child_flat_idx, child_num, BATCH * KFAN, NC,
                                            gamma_c, beta_c, scale_c, shift_c);

    root_kernel<<<BATCH / 8, 256, 0, stream>>>(idx, root_cat, root_num, emb_root,
                                               scale_r, shift_r, out);

    child_kernel<<<(BATCH * KFAN) / 64, 256, 0, stream>>>(
        child_flat_idx, child_cat, child_num, emb_child,
        Wmat, bias, scale_c, shift_c, out);
}